// BottomRightPool_54357106098213
// MI455X (gfx1250) — compile-verified
//
#include <hip/hip_runtime.h>
#include <hip/hip_bf16.h>

// BottomRightPool: out[b,c,i,j] = cummax over H then W of x (16,256,128,128) f32.
// Pure HBM-bandwidth op (512 MB traffic ~22us @ 23.3 TB/s). One 128x128 plane
// per workgroup; scans done in LDS with a 132-float row stride (bank-conflict
// free for both column-strided and row-vector access). Async global<->LDS b128
// transfers (ASYNCcnt) on the load and store ends.

#define PH 128          // plane height
#define PW 128          // plane width
#define LSTRIDE 132     // 128 + 4 pad floats: column access bank = (4i+t)&63

#if defined(__HIP_DEVICE_COMPILE__) && defined(__gfx1250__)
#if __has_builtin(__builtin_amdgcn_global_load_async_to_lds_b128)
#define HAVE_ASYNC_LOAD 1
#endif
#if __has_builtin(__builtin_amdgcn_global_store_async_from_lds_b128)
#define HAVE_ASYNC_STORE 1
#endif
#endif

typedef int v4i __attribute__((ext_vector_type(4)));
typedef __attribute__((address_space(1))) v4i* g4p;   // global int4*
typedef __attribute__((address_space(3))) v4i* l4p;   // LDS int4*

__global__ __launch_bounds__(128) void brpool_kernel(const float* __restrict__ in,
                                                     float* __restrict__ out) {
    __shared__ float tile[PH * LSTRIDE];   // 67,584 B -> 4 workgroups / WGP

    const int t = threadIdx.x;                       // 0..127
    const size_t base = (size_t)blockIdx.x * (PH * PW);

    // ---------------- Phase 0: global plane -> LDS (coalesced b128) ----------
    {
        const int lane4 = t & 31;     // wave covers one full 512B row
        const int rowo  = t >> 5;     // 4 waves -> 4 rows per iteration
#if HAVE_ASYNC_LOAD
        for (int i = 0; i < PH; i += 4) {
            const int row = i + rowo;
            const float* g = in + base + (size_t)row * PW + lane4 * 4;
            float* l = &tile[row * LSTRIDE + lane4 * 4];
            __builtin_amdgcn_global_load_async_to_lds_b128(
                (g4p)g, (l4p)l, /*offset=*/0, /*cpol=*/0);
        }
        asm volatile("s_wait_asynccnt 0x0" ::: "memory");
#else
        #pragma unroll 8
        for (int i = 0; i < PH; i += 4) {
            const int row = i + rowo;
            float4 v = *(const float4*)(in + base + (size_t)row * PW + lane4 * 4);
            *(float4*)(&tile[row * LSTRIDE + lane4 * 4]) = v;
        }
#endif
    }
    __syncthreads();

    // ------- Phase 1: cummax down columns (H axis); thread t owns column t ---
    {
        float running = -INFINITY;
        #pragma unroll 8
        for (int i = 0; i < PH; ++i) {
            const float v = tile[i * LSTRIDE + t];   // bank (4i+t)&63: conflict-free
            running = fmaxf(running, v);
            tile[i * LSTRIDE + t] = running;
        }
    }
    __syncthreads();

    // ------- Phase 2: cummax along rows (W axis); thread t owns row t --------
    {
        float running = -INFINITY;
        float* rowp = &tile[t * LSTRIDE];
        #pragma unroll 8
        for (int j = 0; j < PW; j += 4) {
            float4 v = *(float4*)(rowp + j);         // 2-way (ideal) b128 LDS
            v.x = running = fmaxf(running, v.x);
            v.y = running = fmaxf(running, v.y);
            v.z = running = fmaxf(running, v.z);
            v.w = running = fmaxf(running, v.w);
            *(float4*)(rowp + j) = v;
        }
    }
    __syncthreads();

    // ---------------- Phase 3: LDS -> global (coalesced b128) ----------------
    {
        const int lane4 = t & 31;
        const int rowo  = t >> 5;
#if HAVE_ASYNC_STORE
        for (int i = 0; i < PH; i += 4) {
            const int row = i + rowo;
            float* g = out + base + (size_t)row * PW + lane4 * 4;
            float* l = &tile[row * LSTRIDE + lane4 * 4];
            __builtin_amdgcn_global_store_async_from_lds_b128(
                (g4p)g, (l4p)l, /*offset=*/0, /*cpol=*/0);
        }
        asm volatile("s_wait_asynccnt 0x0" ::: "memory");
#else
        #pragma unroll 8
        for (int i = 0; i < PH; i += 4) {
            const int row = i + rowo;
            float4 v = *(const float4*)(&tile[row * LSTRIDE + lane4 * 4]);
            *(float4*)(out + base + (size_t)row * PW + lane4 * 4) = v;
        }
#endif
    }
}

extern "C" void kernel_launch(void* const* d_in, const int* in_sizes, int n_in,
                              void* d_out, int out_size, void* d_ws, size_t ws_size,
                              hipStream_t stream) {
    (void)n_in; (void)d_ws; (void)ws_size; (void)out_size;
    const float* x = (const float*)d_in[0];
    float* out = (float*)d_out;
    const int nplanes = in_sizes[0] / (PH * PW);   // 16*256 = 4096
    brpool_kernel<<<nplanes, 128, 0, stream>>>(x, out);
}